// DWOp_31430570672475
// MI455X (gfx1250) — compile-verified
//
#include <hip/hip_runtime.h>

typedef __attribute__((ext_vector_type(2))) float v2f;
typedef __attribute__((ext_vector_type(4))) float v4f;
typedef __attribute__((ext_vector_type(8))) float v8f;

#define ZDIM  64
#define NOUT  2304          // 256 * 3 * 3
#define BATCH 16
#define CH    256
#define HH    128
#define WW    128

// ---------------------------------------------------------------------------
// Kernel 1: hypernetwork GEMM  W_dw[b, n] = sum_k z[b,k] * W_lin[n,k]
//           (16 x 64) @ (64 x 2304) -> (16 x 2304)
// One wave (32 lanes) per 16-wide N tile; K swept in chunks of 4 via
// V_WMMA_F32_16X16X4_F32 (full fp32 precision).
//
// A (16x4 f32, 2 VGPRs): lanes 0-15 -> M=lane, K = k0+0 / k0+1
//                        lanes 16-31 -> M=lane-16, K = k0+2 / k0+3
// B (4x16 f32, 2 VGPRs): same split with N in place of M.
// C/D (16x16 f32, 8 VGPRs): VGPR r, lanes 0-15 -> (M=r,   N=lane)
//                                    lanes 16-31 -> (M=r+8, N=lane-16)
// ---------------------------------------------------------------------------
__global__ __launch_bounds__(32)
void hyper_gemm_wmma(const float* __restrict__ z,
                     const float* __restrict__ Wlin,
                     float* __restrict__ Wdw) {
  const int lane  = threadIdx.x;        // 0..31
  const int nbase = blockIdx.x * 16;    // N tile base (0..2288)
  const int half  = lane >> 4;          // 0 or 1
  const int l15   = lane & 15;

  const float* zrow = z    + l15 * ZDIM;                   // A row: M = l15
  const float* wrow = Wlin + (size_t)(nbase + l15) * ZDIM; // B col: N = l15

  v8f acc = {};
#pragma unroll
  for (int k0 = 0; k0 < ZDIM; k0 += 4) {
    const int kb = k0 + half * 2;
    v2f a, b;
    a.x = zrow[kb];  a.y = zrow[kb + 1];
    b.x = wrow[kb];  b.y = wrow[kb + 1];
    // 8 args: (neg_a, A, neg_b, B, c_mod, C, reuse_a, reuse_b)
    acc = __builtin_amdgcn_wmma_f32_16x16x4_f32(
        /*neg_a=*/false, a, /*neg_b=*/false, b,
        /*c_mod=*/(short)0, acc, /*reuse_a=*/false, /*reuse_b=*/false);
  }

#pragma unroll
  for (int r = 0; r < 8; ++r) {
    const int m = r + half * 8;                      // batch index
    Wdw[(size_t)m * NOUT + nbase + l15] = acc[r];
  }
}

// ---------------------------------------------------------------------------
// Kernel 2: depthwise 3x3 conv, per-sample kernels from d_ws.
// Memory-bound (~2.2 FLOP/B): float4-vectorized along W, coalesced B128
// loads/stores; each block works inside one (b,c) plane so the 3-row
// stencil re-reads hit cache. Output stored non-temporally (never re-read).
// Grid: (B*C) * 16 blocks of 256 threads; thread -> one v4f of output.
// ---------------------------------------------------------------------------
__global__ __launch_bounds__(256)
void dwconv3x3(const float* __restrict__ x,
               const float* __restrict__ Wdw,
               float* __restrict__ out) {
  const int plane = blockIdx.x >> 4;                         // b*256 + c
  const int task  = ((blockIdx.x & 15) << 8) | threadIdx.x;  // 0..4095
  const int w     = (task & 31) * 4;                         // 0..124
  const int h     = task >> 5;                               // 0..127

  const int b = plane >> 8;
  const int c = plane & 255;
  const float* wk = Wdw + (size_t)b * NOUT + c * 9;          // wave-uniform
  const float k00 = wk[0], k01 = wk[1], k02 = wk[2];
  const float k10 = wk[3], k11 = wk[4], k12 = wk[5];
  const float k20 = wk[6], k21 = wk[7], k22 = wk[8];

  const float* base = x + (size_t)plane * (HH * WW);
  v4f acc = {0.f, 0.f, 0.f, 0.f};

  auto accum_row = [&](int r, float ka, float kbm, float kc) {
    if (r < 0 || r >= HH) return;
    const float* rp = base + r * WW;
    const v4f cv = *(const v4f*)(rp + w);           // aligned: w % 4 == 0
    const float lf = (w > 0)       ? rp[w - 1] : 0.f;
    const float rt = (w + 4 < WW)  ? rp[w + 4] : 0.f;
    acc.x += ka * lf   + kbm * cv.x + kc * cv.y;
    acc.y += ka * cv.x + kbm * cv.y + kc * cv.z;
    acc.z += ka * cv.y + kbm * cv.z + kc * cv.w;
    acc.w += ka * cv.z + kbm * cv.w + kc * rt;
  };

  accum_row(h - 1, k00, k01, k02);
  accum_row(h,     k10, k11, k12);
  accum_row(h + 1, k20, k21, k22);

  v4f* op = (v4f*)(out + (size_t)plane * (HH * WW) + h * WW + w);
  __builtin_nontemporal_store(acc, op);
}

// ---------------------------------------------------------------------------
extern "C" void kernel_launch(void* const* d_in, const int* in_sizes, int n_in,
                              void* d_out, int out_size, void* d_ws, size_t ws_size,
                              hipStream_t stream) {
  const float* x    = (const float*)d_in[0];   // (16, 256, 128, 128)
  const float* z    = (const float*)d_in[1];   // (16, 64)
  const float* Wlin = (const float*)d_in[2];   // (2304, 64)
  float* out = (float*)d_out;                  // (16, 256, 128, 128)
  float* Wdw = (float*)d_ws;                   // needs 16*2304*4 = 147456 B

  hyper_gemm_wmma<<<NOUT / 16, 32, 0, stream>>>(z, Wlin, Wdw);
  dwconv3x3<<<BATCH * CH * 16, 256, 0, stream>>>(x, Wdw, out);
}